// RNN_9311489098462
// MI455X (gfx1250) — compile-verified
//
#include <hip/hip_runtime.h>
#include <math.h>
#include <stdint.h>

typedef __attribute__((ext_vector_type(2))) float v2f;
typedef __attribute__((ext_vector_type(8))) float v8f;

#define INPUT_SZ   15
#define HIDDEN_SZ  20
#define TB         8                       // timesteps staged per LDS block
#define XS_TSTRIDE 16                      // floats per timestep in LDS (15 + zero pad)
#define XS_ROWSTR  (TB * XS_TSTRIDE + 2)   // 130 floats: even -> b64-aligned, bank-conflict-free
#define XS_BUF     (16 * XS_ROWSTR)        // one staging buffer (floats)
#define HS_ROWSTR  22                      // 20 + 2 pad: even -> b64-aligned, conflict-free
#define WAVES_PER_BLOCK 2
#define LDS_PER_WAVE (2 * XS_BUF + 16 * HS_ROWSTR)

static __device__ __forceinline__ v8f wmma4(v2f a, v2f b, v8f c) {
  // D = A(16x4 f32) * B(4x16 f32) + C(16x16 f32)
  return __builtin_amdgcn_wmma_f32_16x16x4_f32(
      /*neg_a=*/false, a, /*neg_b=*/false, b,
      /*c_mod=*/(short)0, c, /*reuse_a=*/false, /*reuse_b=*/false);
}

static __device__ __forceinline__ float fast_tanh(float x) {
#if __has_builtin(__builtin_amdgcn_tanhf)
  return __builtin_amdgcn_tanhf(x);
#elif __has_builtin(__builtin_amdgcn_tanh_f32)
  return __builtin_amdgcn_tanh_f32(x);
#else
  float y;
  // v_tanh_f32 is a TRANS op: pad with v_nop before the result is consumed.
  asm volatile("v_tanh_f32 %0, %1\n\tv_nop\n\tv_nop" : "=v"(y) : "v"(x));
  return y;
#endif
}

// One dword, global -> LDS, no VGPR landing. GVS mode: mem = SADDR64 + VADDR32.
static __device__ __forceinline__ void async_b32_to_lds(uint32_t lds_byte_addr,
                                                        uint32_t global_byte_off,
                                                        const float* sbase) {
  asm volatile("global_load_async_to_lds_b32 %0, %1, %2"
               :
               : "v"(lds_byte_addr), "v"(global_byte_off), "s"(sbase)
               : "memory");
}

static __device__ __forceinline__ void wait_async0() {
  asm volatile("s_wait_asynccnt 0x0" ::: "memory");
}

__global__ __launch_bounds__(32 * WAVES_PER_BLOCK)
void rnn_fused_wmma(const float* __restrict__ x,    // [B, T, 15]
                    const float* __restrict__ Wih,  // [20, 15]
                    const float* __restrict__ Whh,  // [20, 20]
                    const float* __restrict__ bih,  // [20]
                    const float* __restrict__ bhh,  // [20]
                    float* __restrict__ out,        // [B, 20]
                    int Bsz, int Tsz) {
  __shared__ float lds[WAVES_PER_BLOCK * LDS_PER_WAVE];

  const int lane = threadIdx.x & 31;
  const int wave = threadIdx.x >> 5;
  float* xs = lds + wave * LDS_PER_WAVE;      // [2][16][XS_ROWSTR] double-buffered x stage
  float* hs = xs + 2 * XS_BUF;                // [16][HS_ROWSTR] hidden state

  const int tile = blockIdx.x * WAVES_PER_BLOCK + wave;
  const int m0 = tile * 16;                   // first batch row of this wave's tile
  if (m0 >= Bsz) return;

  const int nlo   = lane & 15;                // column within an N-tile
  const int khalf = (lane >> 4) << 1;         // 0 for lanes 0-15, 2 for lanes 16-31
  const int mrow  = lane & 15;                // A-matrix row this lane feeds

  // ---- hoist weight B-fragments into VGPRs (loop-invariant) -----------------
  // B-fragment layout (4x16 KxN, 2 VGPRs): vgpr j holds B[k = khalf + j][n = nlo]
  v2f wihf[2][4];   // [ntile][kchunk], zero-padded k>=15 / n>=20
  v2f whhf[2][5];   // [ntile][kchunk], K = 20 exactly = 5*4
#pragma unroll
  for (int nt = 0; nt < 2; ++nt) {
    const int n = nt * 16 + nlo;
#pragma unroll
    for (int c = 0; c < 4; ++c) {
      const int k0 = 4 * c + khalf;
      wihf[nt][c][0] = (n < HIDDEN_SZ && k0 + 0 < INPUT_SZ) ? Wih[n * INPUT_SZ + k0 + 0] : 0.f;
      wihf[nt][c][1] = (n < HIDDEN_SZ && k0 + 1 < INPUT_SZ) ? Wih[n * INPUT_SZ + k0 + 1] : 0.f;
    }
#pragma unroll
    for (int c = 0; c < 5; ++c) {
      const int k0 = 4 * c + khalf;
      whhf[nt][c][0] = (n < HIDDEN_SZ) ? Whh[n * HIDDEN_SZ + k0 + 0] : 0.f;
      whhf[nt][c][1] = (n < HIDDEN_SZ) ? Whh[n * HIDDEN_SZ + k0 + 1] : 0.f;
    }
  }
  // bias enters the accumulator as the C operand of the first WMMA chunk
  v8f bias0v, bias1v;
  {
    const float b0 = bih[nlo] + bhh[nlo];
    const float b1 = (nlo < 4) ? (bih[16 + nlo] + bhh[16 + nlo]) : 0.f;
#pragma unroll
    for (int i = 0; i < 8; ++i) { bias0v[i] = b0; bias1v[i] = b1; }
  }

  // ---- init LDS: h = 0, and the k==15 zero-pad column of both x buffers -----
  for (int i = lane; i < 16 * HS_ROWSTR; i += 32) hs[i] = 0.f;
  for (int i = lane; i < 2 * 16 * TB; i += 32) {
    const int buf = i >> 7, rem = i & 127;
    const int r = rem >> 3, t = rem & (TB - 1);
    xs[buf * XS_BUF + r * XS_ROWSTR + t * XS_TSTRIDE + INPUT_SZ] = 0.f;
  }

  const uint32_t rowstride_e = (uint32_t)(Tsz * INPUT_SZ);   // elements

  // ---- async stage of one TB-block of x into buffer `buf` -------------------
  auto stage = [&](int buf, int tb) {
    const uint32_t ldsbase = (uint32_t)(uintptr_t)(xs + buf * XS_BUF);
    for (int f = lane; f < 16 * TB * INPUT_SZ; f += 32) {
      const int row = f / (TB * INPUT_SZ);
      const int pos = f - row * (TB * INPUT_SZ);   // contiguous within the row
      const int t = pos / INPUT_SZ;
      const int k = pos - t * INPUT_SZ;
      const uint32_t laddr = ldsbase + 4u * (uint32_t)(row * XS_ROWSTR + t * XS_TSTRIDE + k);
      const uint32_t goff  = 4u * ((uint32_t)(m0 + row) * rowstride_e +
                                   (uint32_t)(tb * INPUT_SZ + pos));
      async_b32_to_lds(laddr, goff, x);
    }
  };

  stage(0, 0);
  int bufsel = 0;

  for (int tb = 0; tb < Tsz; tb += TB) {
    wait_async0();                       // current buffer resident in LDS
    if (tb + TB < Tsz) stage(bufsel ^ 1, tb + TB);   // DMA next block during compute

    const float* xsb = xs + bufsel * XS_BUF;

    for (int tl = 0; tl < TB; ++tl) {
      v8f acc0, acc1;

      // x_t @ Wih^T : K = 16 (15 real + zero pad), 4 chunks; chunk 0 carries bias as C
      const float* xrow = xsb + mrow * XS_ROWSTR + tl * XS_TSTRIDE;
      {
        v2f a = *(const v2f*)(xrow + khalf);
        acc0 = wmma4(a, wihf[0][0], bias0v);
        acc1 = wmma4(a, wihf[1][0], bias1v);
      }
#pragma unroll
      for (int c = 1; c < 4; ++c) {
        v2f a = *(const v2f*)(xrow + 4 * c + khalf);   // ds_load_b64, conflict-free
        acc0 = wmma4(a, wihf[0][c], acc0);
        acc1 = wmma4(a, wihf[1][c], acc1);
      }
      // h_{t-1} @ Whh^T : K = 20, 5 chunks
      const float* hrow = hs + mrow * HS_ROWSTR;
#pragma unroll
      for (int c = 0; c < 5; ++c) {
        v2f a = *(const v2f*)(hrow + 4 * c + khalf);
        acc0 = wmma4(a, whhf[0][c], acc0);
        acc1 = wmma4(a, whhf[1][c], acc1);
      }

      // hardware tanh + write h back to LDS in C-layout scatter (conflict-free)
      const int mbase = (lane >> 4) << 3;              // rows 0..7 or 8..15
#pragma unroll
      for (int r = 0; r < 8; ++r)
        hs[(mbase + r) * HS_ROWSTR + nlo] = fast_tanh(acc0[r]);
      if (nlo < 4) {
#pragma unroll
        for (int r = 0; r < 8; ++r)
          hs[(mbase + r) * HS_ROWSTR + 16 + nlo] = fast_tanh(acc1[r]);
      }
    }
    bufsel ^= 1;
  }

  // ---- final hidden state: 320 contiguous floats per wave, coalesced --------
  for (int f = lane; f < 16 * HIDDEN_SZ; f += 32) {
    const int row = f / HIDDEN_SZ;
    const int n = f - row * HIDDEN_SZ;
    out[(size_t)m0 * HIDDEN_SZ + f] = hs[row * HS_ROWSTR + n];
  }
}

extern "C" void kernel_launch(void* const* d_in, const int* in_sizes, int n_in,
                              void* d_out, int out_size, void* d_ws, size_t ws_size,
                              hipStream_t stream) {
  const float* feature = (const float*)d_in[0];  // [B, T, 15]
  const float* W_ih    = (const float*)d_in[1];  // [20, 15]
  const float* W_hh    = (const float*)d_in[2];  // [20, 20]
  const float* b_ih    = (const float*)d_in[3];  // [20]
  const float* b_hh    = (const float*)d_in[4];  // [20]
  float* out = (float*)d_out;                    // [B, 20]

  const int B = out_size / HIDDEN_SZ;                       // 4096
  const int T = in_sizes[0] / (B * INPUT_SZ);               // 512

  const int tiles = (B + 15) / 16;                          // 256
  const int blocks = (tiles + WAVES_PER_BLOCK - 1) / WAVES_PER_BLOCK;  // 128
  rnn_fused_wmma<<<blocks, 32 * WAVES_PER_BLOCK, 0, stream>>>(
      feature, W_ih, W_hh, b_ih, b_hh, out, B, T);
}